// CRFLoss_vb_PA_47382079209904
// MI455X (gfx1250) — compile-verified
//
#include <hip/hip_runtime.h>
#include <hip/hip_bf16.h>

// CRF forward loss for MI455X (gfx1250).
// Inputs (setup_inputs order): scores f32 [128,256,48,48], target bool [128,256,48],
// mask bool [128,256]. Bool arrays assumed 1 byte per element (numpy/jax bool_).
// Output: single f32 scalar loss.
//
// Design: 1 workgroup per batch row; 192 threads = 6 wave32s; 4 lanes per tag
// column, interleaved row assignment (bank-conflict-free tile reads); score
// tile double-buffered in LDS and streamed with global_load_async_to_lds_b128
// (ASYNCcnt); target/mask staged in LDS once; double-buffered state vectors
// give one workgroup barrier per timestep.

constexpr int kTags  = 48;
constexpr int kSeq   = 256;
constexpr int kBatch = 128;
constexpr int kStart = 46;
constexpr int kEnd   = 47;
constexpr float kNinf = -100000.0f;
constexpr int kTile  = kTags * kTags;            // 2304 floats = 9216 B per step
constexpr int kThreads = 192;                    // 6 wave32s; 4 lanes per column
constexpr int kRowsPerLane = kTags / 4;          // 12
constexpr int kTileChunks = (kTile * 4 / 16) / kThreads;      // 576/192 = 3
constexpr int kTgtChunks  = (kSeq * kTags / 16) / kThreads;   // 768/192 = 4

#ifndef __has_builtin
#define __has_builtin(x) 0
#endif

#if __has_builtin(__builtin_amdgcn_global_load_async_to_lds_b128) && \
    __has_builtin(__builtin_amdgcn_s_wait_asynccnt)
#define USE_ASYNC_LDS 1
#else
#define USE_ASYNC_LDS 0
#endif

typedef int v4i __attribute__((ext_vector_type(4)));
typedef __attribute__((address_space(1))) v4i* gv4i_p;   // global (AS1)
typedef __attribute__((address_space(3))) v4i* lv4i_p;   // LDS (AS3)

__device__ __forceinline__ void async_copy16(const void* g, void* l) {
#if USE_ASYNC_LDS
  // 16B per lane, DMA global -> LDS, tracked by ASYNCcnt (no VGPR data path).
  __builtin_amdgcn_global_load_async_to_lds_b128(
      (gv4i_p)(g), (lv4i_p)(l), /*offset=*/0, /*cpol=*/0);
#else
  *(float4*)l = *(const float4*)g;   // synchronous fallback
#endif
}

__device__ __forceinline__ void wait_async_all() {
#if USE_ASYNC_LDS
  __builtin_amdgcn_s_wait_asynccnt(0);
#endif
}

__global__ __launch_bounds__(kThreads)
void crf_fwd_kernel(const float* __restrict__ scores,
                    const unsigned char* __restrict__ target,
                    const unsigned char* __restrict__ mask,
                    float* __restrict__ contrib) {
  const int b   = blockIdx.x;
  const int tid = threadIdx.x;        // 0..191
  const int j   = tid >> 2;           // tag column 0..47
  const int q   = tid & 3;            // reduction quarter; rows i = q + 4u

  const float*         gs = scores + (size_t)b * kSeq * kTile;
  const unsigned char* tg = target + (size_t)b * kSeq * kTags;
  const unsigned char* mk = mask   + (size_t)b * kSeq;

  __shared__ __align__(16) float buf[2][kTile];            // score tile, double buffer
  __shared__ __align__(16) unsigned char tgl[kSeq * kTags]; // all targets for this row
  __shared__ __align__(16) unsigned char mkl[kSeq];         // all mask bytes
  __shared__ float pstate[2][kTags];                        // partition, double buffer
  __shared__ float tstate[2][kTags];                        // tag_partition, double buffer

  // ---- t = 0 init: partition = scores[b,0,START,:]; tag_partition masked by target[0]
  if (tid < kTags) {
    float iv = gs[kStart * kTags + tid];
    pstate[0][tid] = iv;
    tstate[0][tid] = (tg[tid] != 0) ? kNinf : iv;
  }

  // ---- prologue: async-stage target (12 KB), mask (256 B), and tile t=1
#pragma unroll
  for (int k = 0; k < kTgtChunks; ++k) {
    const int c = k * kThreads + tid;                       // 16B chunk index
    async_copy16(tg + c * 16, tgl + c * 16);
  }
  if (tid < kSeq / 16) {
    async_copy16(mk + tid * 16, mkl + tid * 16);
  }
  {
    const float* gsrc = gs + (size_t)1 * kTile;
#pragma unroll
    for (int k = 0; k < kTileChunks; ++k) {
      const int c = k * kThreads + tid;
      async_copy16(gsrc + c * 4, buf[1] + c * 4);
    }
  }

  for (int t = 1; t < kSeq; ++t) {
    wait_async_all();                 // this wave's in-flight DMA (tile t, +staging) done
    __syncthreads();                  // all waves' DMA landed; state(t-1) writes visible

    // Prefetch tile t+1. Safe after the barrier: buf[(t+1)&1] held tile t-1,
    // fully consumed before barrier(t).
    if (t < kSeq - 1) {
      const float* gsrc = gs + (size_t)(t + 1) * kTile;
      float* ld = buf[(t + 1) & 1];
#pragma unroll
      for (int k = 0; k < kTileChunks; ++k) {
        const int c = k * kThreads + tid;
        async_copy16(gsrc + c * 4, ld + c * 4);
      }
    }

    const float* tile = buf[t & 1];
    const float* pr = pstate[(t - 1) & 1];
    const float* tr = tstate[(t - 1) & 1];

    // Cache this lane's 12 interleaved rows of column j (bank-conflict-free:
    // word addr = (q + 4u)*48 + j -> banks {0,48,32,16}+j mod 64, all distinct).
    float sv[kRowsPerLane];
#pragma unroll
    for (int u = 0; u < kRowsPerLane; ++u) sv[u] = tile[(q + 4 * u) * kTags + j];

    // Pass 1: column max for both recurrences (numerically stable LSE).
    float mxp = -3.0e38f, mxt = -3.0e38f;
#pragma unroll
    for (int u = 0; u < kRowsPerLane; ++u) {
      const int i = q + 4 * u;
      mxp = fmaxf(mxp, sv[u] + pr[i]);
      mxt = fmaxf(mxt, sv[u] + tr[i]);
    }
    // Combine across the 4 lanes of this column (always within one wave32).
    mxp = fmaxf(mxp, __shfl_xor(mxp, 1, 32));
    mxt = fmaxf(mxt, __shfl_xor(mxt, 1, 32));
    mxp = fmaxf(mxp, __shfl_xor(mxp, 2, 32));
    mxt = fmaxf(mxt, __shfl_xor(mxt, 2, 32));

    // Pass 2: sum of exp.
    float sp = 0.0f, st = 0.0f;
#pragma unroll
    for (int u = 0; u < kRowsPerLane; ++u) {
      const int i = q + 4 * u;
      sp += __expf(sv[u] + pr[i] - mxp);
      st += __expf(sv[u] + tr[i] - mxt);
    }
    sp += __shfl_xor(sp, 1, 32);
    st += __shfl_xor(st, 1, 32);
    sp += __shfl_xor(sp, 2, 32);
    st += __shfl_xor(st, 2, 32);

    const float newp = mxp + __logf(sp);
    const bool  tgt  = tgl[t * kTags + j] != 0;
    const float newt = tgt ? kNinf : (mxt + __logf(st));
    const bool  mv   = mkl[t] != 0;

    // Write state(t) into the other buffer; no trailing barrier needed.
    if (q == 0) {
      float wp = newp, wt = newt;
      if (!mv) { wp = pr[j]; wt = tr[j]; }   // carry forward past sequence end
      pstate[t & 1][j] = wp;
      tstate[t & 1][j] = wt;
    }
  }

  __syncthreads();
  if (tid == 0) {
    const int fin = (kSeq - 1) & 1;   // final state buffer
    const float pe = pstate[fin][kEnd];
    float te = tstate[fin][kEnd];
    te = (te == kNinf) ? 0.0f : te;   // reference: where(tp == NINF, 0, tp)
    contrib[b] = pe - te;             // loss contribution of this batch row
  }
}

__global__ void crf_reduce_kernel(const float* __restrict__ contrib,
                                  float* __restrict__ out) {
  if (threadIdx.x == 0 && blockIdx.x == 0) {
    float s = 0.0f;
    for (int i = 0; i < kBatch; ++i) s += contrib[i];  // fixed order: deterministic
    out[0] = s;
  }
}

extern "C" void kernel_launch(void* const* d_in, const int* in_sizes, int n_in,
                              void* d_out, int out_size, void* d_ws, size_t ws_size,
                              hipStream_t stream) {
  (void)in_sizes; (void)n_in; (void)out_size; (void)ws_size;
  const float*         scores = (const float*)d_in[0];
  const unsigned char* target = (const unsigned char*)d_in[1];
  const unsigned char* mask   = (const unsigned char*)d_in[2];
  float* contrib = (float*)d_ws;     // 128 floats of scratch
  float* out     = (float*)d_out;

  crf_fwd_kernel<<<kBatch, kThreads, 0, stream>>>(scores, target, mask, contrib);
  crf_reduce_kernel<<<1, 32, 0, stream>>>(contrib, out);
}